// GAT_27994596836122
// MI455X (gfx1250) — compile-verified
//
#include <hip/hip_runtime.h>
#include <math.h>

typedef __attribute__((ext_vector_type(2))) float v2f;
typedef __attribute__((ext_vector_type(8))) float v8f;

#define NEG_SLOPE 0.2f
#define GAT_EPS   1e-16f
#define FLT_BIG   3.402823466e+38f

// ---------------------------------------------------------------------------
// fill
// ---------------------------------------------------------------------------
__global__ void k_fill(float* __restrict__ p, float v, int count) {
  int i = blockIdx.x * blockDim.x + threadIdx.x;
  if (i < count) p[i] = v;
}

// ---------------------------------------------------------------------------
// WMMA fp32 GEMM:  H[n x WCOLS] = X[n x K] * W[K x WCOLS]
// Compile-time K / LDX / LDW / WCOLS / NT -> fully unrolled, immediate-offset
// loads, no divergent guards (indices clamped; tail rows/cols never stored).
//
// One wave per 16-row stripe; the wave computes all NT 16-col tiles with NT
// accumulators so each A fragment is reused NT times.
// A operand (16x4 f32, 2 VGPR): lanes 0-15 -> M=lane,    K = k+0,k+1
//                               lanes16-31 -> M=lane-16, K = k+2,k+3
// B operand (4x16 f32, 2 VGPR): same K split across lane halves.
// C/D (16x16 f32, 8 VGPR): vgpr j, lanes0-15: M=j, N=lane; lanes16-31: M=j+8
// ---------------------------------------------------------------------------
template <int K, int LDX, int LDW, int WCOLS, int NT>
__global__ void k_gemm_wmma(const float* __restrict__ X, const float* __restrict__ W,
                            float* __restrict__ H, int n) {
  const int wave = threadIdx.x >> 5;               // 4 waves per block
  const int rowBase = (blockIdx.x * 4 + wave) * 16;
  if (rowBase >= n) return;
  const int lane = threadIdx.x & 31;
  const int half = lane >> 4;                      // 0: K+0/1   1: K+2/3
  const int l15  = lane & 15;

  int aRow = rowBase + l15;
  if (aRow >= n) aRow = n - 1;                     // clamp: tail never stored
  const float* aPtr = X + (size_t)aRow * LDX + half * 2;

  const float* bPtr[NT];
  v8f acc[NT];
#pragma unroll
  for (int t = 0; t < NT; ++t) {
    int c = t * 16 + l15;
    if (c >= WCOLS) c = WCOLS - 1;                 // clamp: tail never stored
    bPtr[t] = W + (size_t)(half * 2) * LDW + c;
    acc[t] = (v8f){};
  }

#pragma unroll
  for (int k = 0; k < K; k += 4) {
    const v2f a = *(const v2f*)(aPtr + k);         // global_load_b64, imm offs
#pragma unroll
    for (int t = 0; t < NT; ++t) {
      v2f b;
      b.x = bPtr[t][(size_t)k * LDW];
      b.y = bPtr[t][(size_t)(k + 1) * LDW];
      acc[t] = __builtin_amdgcn_wmma_f32_16x16x4_f32(
          false, a, false, b, (short)0, acc[t], false, false);
    }
  }

  const bool rowsFull = (rowBase + 16 <= n);
#pragma unroll
  for (int t = 0; t < NT; ++t) {
    const int col = t * 16 + l15;
    if ((t + 1) * 16 > WCOLS && col >= WCOLS) continue;   // partial tile only
#pragma unroll
    for (int j = 0; j < 8; ++j) {
      const int row = rowBase + j + half * 8;
      if (rowsFull || row < n)
        H[(size_t)row * WCOLS + col] = acc[t][j];
    }
  }
}

// ---------------------------------------------------------------------------
// per-(node,head) attention scores:  a_src[n,Hh], a_dst[n,Hh]
// ---------------------------------------------------------------------------
__global__ void k_att_scores(const float* __restrict__ Hm,
                             const float* __restrict__ att_s,
                             const float* __restrict__ att_d,
                             float* __restrict__ as_, float* __restrict__ ad_,
                             int n, int Hh, int C) {
  int i = blockIdx.x * blockDim.x + threadIdx.x;
  if (i >= n * Hh) return;
  int h = i % Hh;
  int node = i / Hh;
  const float* hp = Hm + (size_t)node * Hh * C + (size_t)h * C;
  float s = 0.f, d = 0.f;
  for (int c = 0; c < C; ++c) {
    float v = hp[c];
    s += v * att_s[h * C + c];
    d += v * att_d[h * C + c];
  }
  as_[i] = s;
  ad_[i] = d;
}

// ---------------------------------------------------------------------------
// float atomic max via CAS (emax initialized to -FLT_BIG, never NaN)
// ---------------------------------------------------------------------------
__device__ __forceinline__ void atomicMaxF32(float* addr, float val) {
  int* ia = (int*)addr;
  int cur = __float_as_int(*addr);
  while (__int_as_float(cur) < val) {
    int assumed = cur;
    cur = atomicCAS(ia, assumed, __float_as_int(val));
    if (cur == assumed) break;
  }
}

// edge pass 1: segment-max of leaky_relu(a_src[src]+a_dst[dst]) per dst/head
__global__ void k_edge_max(const int* __restrict__ src, const int* __restrict__ dst,
                           int Ereal, int n,
                           const float* __restrict__ as_, const float* __restrict__ ad_,
                           float* __restrict__ emax, int Hh) {
  int e = blockIdx.x * blockDim.x + threadIdx.x;
  int total = Ereal + n;
  if (e >= total) return;
  int s = (e < Ereal) ? src[e] : (e - Ereal);   // self-loops appended
  int d = (e < Ereal) ? dst[e] : (e - Ereal);
  for (int h = 0; h < Hh; ++h) {
    float v = as_[s * Hh + h] + ad_[d * Hh + h];
    v = (v > 0.f) ? v : NEG_SLOPE * v;
    atomicMaxF32(&emax[d * Hh + h], v);
  }
}

// ---------------------------------------------------------------------------
// edge pass 2, layer 1 (H=8, C=8): warp per edge, lane = h*8+c (coalesced).
// num[dst,64] += h1[src,64]*ex ; den[dst,8] += ex
// ---------------------------------------------------------------------------
__global__ void k_edge_agg1(const int* __restrict__ src, const int* __restrict__ dst,
                            int Ereal, int n,
                            const float* __restrict__ as_, const float* __restrict__ ad_,
                            const float* __restrict__ emax,
                            const float* __restrict__ Hm,
                            float* __restrict__ num, float* __restrict__ den) {
  int warp = (blockIdx.x * blockDim.x + threadIdx.x) >> 5;
  int lane = threadIdx.x & 31;
  int total = Ereal + n;
  if (warp >= total) return;
  int s = (warp < Ereal) ? src[warp] : (warp - Ereal);
  int d = (warp < Ereal) ? dst[warp] : (warp - Ereal);
  int c = lane & 7;
#pragma unroll
  for (int halfIter = 0; halfIter < 2; ++halfIter) {
    int h = (lane >> 3) + halfIter * 4;          // 0..3 then 4..7
    float e = as_[s * 8 + h] + ad_[d * 8 + h];
    e = (e > 0.f) ? e : NEG_SLOPE * e;
    float ex = __expf(e - emax[d * 8 + h]);
    atomicAdd(&num[(size_t)d * 64 + h * 8 + c],
              Hm[(size_t)s * 64 + h * 8 + c] * ex);
    if (c == 0) atomicAdd(&den[d * 8 + h], ex);
  }
}

// node pass layer 1: emb = elu(num/(den+eps) + b1)   (concat across 8 heads)
__global__ void k_node_out1(const float* __restrict__ num, const float* __restrict__ den,
                            const float* __restrict__ b1, float* __restrict__ emb, int n) {
  int i = blockIdx.x * blockDim.x + threadIdx.x;   // n*64
  if (i >= n * 64) return;
  int node = i >> 6;
  int f = i & 63;
  int h = f >> 3;
  float v = num[i] / (den[node * 8 + h] + GAT_EPS) + b1[f];
  emb[i] = (v > 0.f) ? v : (__expf(v) - 1.f);      // elu (alpha=1)
}

// ---------------------------------------------------------------------------
// edge pass 2, layer 2 (H=1, C=40): warp per edge.
// ---------------------------------------------------------------------------
__global__ void k_edge_agg2(const int* __restrict__ src, const int* __restrict__ dst,
                            int Ereal, int n,
                            const float* __restrict__ as_, const float* __restrict__ ad_,
                            const float* __restrict__ emax,
                            const float* __restrict__ Hm,
                            float* __restrict__ num, float* __restrict__ den) {
  int warp = (blockIdx.x * blockDim.x + threadIdx.x) >> 5;
  int lane = threadIdx.x & 31;
  int total = Ereal + n;
  if (warp >= total) return;
  int s = (warp < Ereal) ? src[warp] : (warp - Ereal);
  int d = (warp < Ereal) ? dst[warp] : (warp - Ereal);
  float e = as_[s] + ad_[d];
  e = (e > 0.f) ? e : NEG_SLOPE * e;
  float ex = __expf(e - emax[d]);
  for (int c = lane; c < 40; c += 32)
    atomicAdd(&num[(size_t)d * 40 + c], Hm[(size_t)s * 40 + c] * ex);
  if (lane == 0) atomicAdd(&den[d], ex);
}

// node pass layer 2: logits = num/(den+eps) + b2 ; then in-place log_softmax
__global__ void k_node_out2(float* __restrict__ logits, const float* __restrict__ den,
                            const float* __restrict__ b2, int n) {
  int i = blockIdx.x * blockDim.x + threadIdx.x;
  if (i >= n) return;
  float inv = 1.f / (den[i] + GAT_EPS);
  float row[40];
  float mx = -FLT_BIG;
  float* p = logits + (size_t)i * 40;
#pragma unroll
  for (int c = 0; c < 40; ++c) {
    float v = p[c] * inv + b2[c];
    row[c] = v;
    mx = fmaxf(mx, v);
  }
  float sum = 0.f;
#pragma unroll
  for (int c = 0; c < 40; ++c) sum += __expf(row[c] - mx);
  float lse = mx + __logf(sum);
#pragma unroll
  for (int c = 0; c < 40; ++c) p[c] = row[c] - lse;
}

// ---------------------------------------------------------------------------
// launch
// ---------------------------------------------------------------------------
#define CDIV(a, b) (((a) + (b) - 1) / (b))

extern "C" void kernel_launch(void* const* d_in, const int* in_sizes, int n_in,
                              void* d_out, int out_size, void* d_ws, size_t ws_size,
                              hipStream_t stream) {
  const float* x    = (const float*)d_in[0];
  const float* W1   = (const float*)d_in[1];
  const float* aS1w = (const float*)d_in[2];   // att_src1 [8,8]
  const float* aD1w = (const float*)d_in[3];   // att_dst1 [8,8]
  const float* b1   = (const float*)d_in[4];   // [64]
  const float* W2   = (const float*)d_in[5];   // [64,40]
  const float* aS2w = (const float*)d_in[6];   // att_src2 [1,40]
  const float* aD2w = (const float*)d_in[7];   // att_dst2 [1,40]
  const float* b2   = (const float*)d_in[8];   // [40]
  const int*   ei   = (const int*)d_in[9];     // [2,E]

  const int n     = in_sizes[0] / 128;
  const int Ereal = in_sizes[9] / 2;
  const int* srcIdx = ei;
  const int* dstIdx = ei + Ereal;
  const int total   = Ereal + n;               // with self-loops

  float* ws = (float*)d_ws;
  float* h1    = ws; ws += (size_t)n * 64;
  float* aS1   = ws; ws += (size_t)n * 8;
  float* aD1   = ws; ws += (size_t)n * 8;
  float* emax1 = ws; ws += (size_t)n * 8;
  float* den1  = ws; ws += (size_t)n * 8;
  float* num1  = ws; ws += (size_t)n * 64;
  float* h2    = ws; ws += (size_t)n * 40;
  float* aS2   = ws; ws += (size_t)n;
  float* aD2   = ws; ws += (size_t)n;
  float* emax2 = ws; ws += (size_t)n;
  float* den2  = ws; ws += (size_t)n;

  float* logits = (float*)d_out;                    // output 0: [n,40]
  float* emb    = (float*)d_out + (size_t)n * 40;   // output 1: [n,64]

  const int TB = 256;

  // init accumulators (every call: deterministic result)
  k_fill<<<CDIV(n * 8, TB),  TB, 0, stream>>>(emax1, -FLT_BIG, n * 8);
  k_fill<<<CDIV(n * 8, TB),  TB, 0, stream>>>(den1, 0.f, n * 8);
  k_fill<<<CDIV(n * 64, TB), TB, 0, stream>>>(num1, 0.f, n * 64);
  k_fill<<<CDIV(n, TB),      TB, 0, stream>>>(emax2, -FLT_BIG, n);
  k_fill<<<CDIV(n, TB),      TB, 0, stream>>>(den2, 0.f, n);
  k_fill<<<CDIV(n * 40, TB), TB, 0, stream>>>(logits, 0.f, n * 40);

  const int rowTiles = CDIV(n, 16);

  // ---- layer 1 ----
  k_gemm_wmma<128, 128, 64, 64, 4>
      <<<CDIV(rowTiles, 4), 128, 0, stream>>>(x, W1, h1, n);
  k_att_scores<<<CDIV(n * 8, TB), TB, 0, stream>>>(h1, aS1w, aD1w, aS1, aD1, n, 8, 8);
  k_edge_max<<<CDIV(total, TB), TB, 0, stream>>>(srcIdx, dstIdx, Ereal, n, aS1, aD1, emax1, 8);
  k_edge_agg1<<<CDIV(total * 32, TB), TB, 0, stream>>>(srcIdx, dstIdx, Ereal, n,
                                                       aS1, aD1, emax1, h1, num1, den1);
  k_node_out1<<<CDIV(n * 64, TB), TB, 0, stream>>>(num1, den1, b1, emb, n);

  // ---- layer 2 ----
  k_gemm_wmma<64, 64, 40, 40, 3>
      <<<CDIV(rowTiles, 4), 128, 0, stream>>>(emb, W2, h2, n);
  k_att_scores<<<CDIV(n, TB), TB, 0, stream>>>(h2, aS2w, aD2w, aS2, aD2, n, 1, 40);
  k_edge_max<<<CDIV(total, TB), TB, 0, stream>>>(srcIdx, dstIdx, Ereal, n, aS2, aD2, emax2, 1);
  k_edge_agg2<<<CDIV(total * 32, TB), TB, 0, stream>>>(srcIdx, dstIdx, Ereal, n,
                                                       aS2, aD2, emax2, h2, logits, den2);
  k_node_out2<<<CDIV(n, TB), TB, 0, stream>>>(logits, den2, b2, n);

  (void)n_in; (void)out_size; (void)ws_size;
}